// RegularGrid_27599459844803
// MI455X (gfx1250) — compile-verified
//
#include <hip/hip_runtime.h>
#include <math.h>

// ---------------- problem constants (from reference) ----------------
#define RES       128
#define NV        (RES * RES * RES)        // 2,097,152 voxels
#define DATA_DIM  28                       // 27 SH coeffs + 1 sigma
#define NRAYS     4096
#define NSAMP     444                      // ceil(sqrt(3)*2*128)
#define NBLK      14                       // ceil(444/32) wave32 blocks per ray

#define F_RADIUS  1.3f
#define F_VOXEL   (2.0f * F_RADIUS / (float)RES)
#define F_STEP    (0.5f * F_VOXEL)

// strides in the transposed [z][y][x][c] layout (float units)
#define VX_STRIDE (DATA_DIM)               // +1 in x
#define VY_STRIDE (RES * DATA_DIM)         // +1 in y
#define VZ_STRIDE (RES * RES * DATA_DIM)   // +1 in z

typedef unsigned int v4u __attribute__((ext_vector_type(4)));
typedef int          v4i __attribute__((ext_vector_type(4)));
typedef int          v8i __attribute__((ext_vector_type(8)));

// Issue a TDM tensor load (descriptor groups 0/1; groups 2/3 zero = 2D tile).
// ROCm 7.2 (clang-22) exposes a 5-arg builtin; amdgpu-toolchain (clang-23) a 6-arg one.
__device__ __forceinline__ void plx_tdm_load(v4u g0, v8i g1) {
    v4i z4 = {0, 0, 0, 0};
#if defined(__clang_major__) && (__clang_major__ >= 23)
    v8i z8 = {0, 0, 0, 0, 0, 0, 0, 0};
    __builtin_amdgcn_tensor_load_to_lds(g0, g1, z4, z4, z8, 0);
#else
    __builtin_amdgcn_tensor_load_to_lds(g0, g1, z4, z4, 0);
#endif
}

// ---------------------------------------------------------------
// Kernel 1: relayout (C,Z,Y,X) -> (Z,Y,X,C) using the Tensor Data
// Mover.  The source tile is a 2D tensor: dim0 = 512 x-contiguous
// voxels of one channel, dim1 = 28 channels at stride NV floats
// (8 MB).  One TENSOR_LOAD_TO_LDS DMAs the 56 KB tile into LDS
// (rows packed back-to-back: channel c at byte offset c*2048), the
// DMA is tracked on TENSORcnt.  Threads then read conflict-free
// (bank = voxel index; row stride 2048B = multiple of 64 banks) and
// store each voxel's 28 channels as one contiguous 112B record
// (7 aligned b128 stores).
// ---------------------------------------------------------------
#define TP_VOX 512                          // voxels per workgroup tile

__global__ __launch_bounds__(256) void plx_transpose_tdm_kernel(
    const float* __restrict__ in, float* __restrict__ out)
{
    __shared__ float tile[DATA_DIM * TP_VOX];        // 56 KB

    const int v0 = blockIdx.x * TP_VOX;              // first voxel of this tile

    if ((threadIdx.x >> 5) == 0) {                   // wave 0 issues the DMA
        unsigned long long ga = (unsigned long long)(const void*)(in + v0);
        // low 32 bits of the generic (flat) address of an LDS object are the
        // group-relative LDS byte offset (ISA 10.2 aperture mapping)
        unsigned lds_base = (unsigned)(size_t)(void*)&tile[0];

        // ---- D# group 0 (ISA 8.3): count | lds_addr | global_addr | type=2 ----
        v4u g0;
        g0.x = 1u;                                        // [1:0] count=1 (valid)
        g0.y = lds_base;                                  // [63:32] lds_addr
        g0.z = (unsigned)(ga & 0xFFFFFFFFull);            // [95:64] global_addr lo
        g0.w = (unsigned)((ga >> 32) & 0x1FFFFFFull)      // [120:96] global_addr hi
             | (2u << 30);                                // [127:126] type=2
        // ---- D# group 1 (ISA 8.4): sizes / strides ----
        v8i g1;
        g1[0] = (int)(2u << 16);                 // wg_mask=0, data_size=2 -> 4 bytes
        g1[1] = (int)((NV & 0xFFFFu) << 16);     // [63:48] tensor_dim0[15:0]
        g1[2] = (int)((NV >> 16) & 0xFFFFu)      // [79:64] tensor_dim0[31:16]
              | (int)(DATA_DIM << 16);           // [95:80] tensor_dim1[15:0] = 28
        g1[3] = (int)(TP_VOX << 16);             // dim1 hi = 0, [127:112] tile_dim0 = 512
        g1[4] = (int)DATA_DIM;                   // tile_dim1 = 28, tile_dim2 = 0 (2D)
        g1[5] = (int)NV;                         // tensor_dim0_stride[31:0] = NV floats
        g1[6] = 0;                               // stride0 hi = 0, stride1 lo = 0
        g1[7] = 0;                               // stride1 hi = 0

        plx_tdm_load(g0, g1);
        __builtin_amdgcn_s_wait_tensorcnt(0);    // DMA complete for this wave
    }
    __syncthreads();                             // make LDS tile visible to all waves

#pragma unroll
    for (int rep = 0; rep < TP_VOX / 256; ++rep) {
        int i = rep * 256 + threadIdx.x;         // voxel within tile
        float vals[DATA_DIM];
#pragma unroll
        for (int c = 0; c < DATA_DIM; ++c)
            vals[c] = tile[c * TP_VOX + i];      // bank-conflict free

        float4* o = (float4*)(out + (size_t)(v0 + i) * DATA_DIM); // 112B, 16B aligned
#pragma unroll
        for (int q = 0; q < 7; ++q)
            o[q] = make_float4(vals[4*q+0], vals[4*q+1], vals[4*q+2], vals[4*q+3]);
    }
}

__device__ __forceinline__ float plx_sigmoid(float x) {
    return 1.0f / (1.0f + expf(-x));
}

// compute clamped grid coords for a point; returns voxel base index
__device__ __forceinline__ int plx_coords(float px, float py, float pz,
                                          float& fx, float& fy, float& fz,
                                          int& ix, int& iy, int& iz)
{
    const float S = 0.5f * 127.0f / F_RADIUS;    // ((p/R)+1)*0.5*127 = p*S + 63.5
    float gx = fminf(fmaxf(fmaf(px, S, 63.5f), 0.0f), 127.0f);
    float gy = fminf(fmaxf(fmaf(py, S, 63.5f), 0.0f), 127.0f);
    float gz = fminf(fmaxf(fmaf(pz, S, 63.5f), 0.0f), 127.0f);
    ix = min((int)gx, 126);                      // gx >= 0 so trunc == floor
    iy = min((int)gy, 126);
    iz = min((int)gz, 126);
    fx = gx - (float)ix;
    fy = gy - (float)iy;
    fz = gz - (float)iz;
    return ((iz * RES + iy) * RES + ix) * DATA_DIM;
}

// ---------------------------------------------------------------
// Kernel 2: one wave32 per ray.  Lane i of each 32-sample block owns
// sample (blk*32 + i): gathers 8 corner records (56 aligned b128
// loads), computes alpha; the sequential cumprod transmittance is a
// wave-level product scan via __shfl_up (wave32).  RGB/acc are
// butterfly-reduced; lane 0 writes the 3 outputs + background.
// ---------------------------------------------------------------
__global__ __launch_bounds__(128) void plx_render_tx_kernel(
    const float* __restrict__ ro, const float* __restrict__ rd,
    const float* __restrict__ grid, float* __restrict__ out)
{
    const int lane = threadIdx.x & 31;
    const int ray  = blockIdx.x * 4 + (threadIdx.x >> 5);

    // ---- per-ray setup (redundant across lanes, cheap) ----
    float ox = ro[3*ray+0], oy = ro[3*ray+1], oz = ro[3*ray+2];
    float dx = rd[3*ray+0], dy = rd[3*ray+1], dz = rd[3*ray+2];

    float ivx = 1.0f/dx, ivy = 1.0f/dy, ivz = 1.0f/dz;
    float t0x = (-F_RADIUS-ox)*ivx, t1x = (F_RADIUS-ox)*ivx;
    float t0y = (-F_RADIUS-oy)*ivy, t1y = (F_RADIUS-oy)*ivy;
    float t0z = (-F_RADIUS-oz)*ivz, t1z = (F_RADIUS-oz)*ivz;
    float tmin = fmaxf(fmaxf(fminf(t0x,t1x), fminf(t0y,t1y)), fminf(t0z,t1z));
    tmin = fmaxf(tmin, 0.0f);
    float tmax = fminf(fminf(fmaxf(t0x,t1x), fmaxf(t0y,t1y)), fmaxf(t0z,t1z));
    bool  hit  = (tmin < tmax);

    float nd  = sqrtf(dx*dx + dy*dy + dz*dz);
    float inn = 1.0f / (nd + 1e-9f);
    float nx = dx*inn, ny = dy*inn, nz = dz*inn;

    // SH deg-2 basis of the (normalized) ray direction
    float m0 = 0.28209479177387814f;
    float m1 = -0.4886025119029199f * ny;
    float m2 =  0.4886025119029199f * nz;
    float m3 = -0.4886025119029199f * nx;
    float m4 =  1.0925484305920792f * nx * ny;
    float m5 = -1.0925484305920792f * ny * nz;
    float m6 =  0.31539156525252005f * (2.0f*nz*nz - nx*nx - ny*ny);
    float m7 = -1.0925484305920792f * nx * nz;
    float m8 =  0.5462742152960396f * (nx*nx - ny*ny);

    float Tc = 1.0f;                 // transmittance carried across blocks
    float aR = 0.0f, aG = 0.0f, aB = 0.0f, aA = 0.0f;

    for (int blk = 0; blk < NBLK; ++blk) {
        int   s = blk * 32 + lane;
        float t = fmaf((float)s, F_STEP, tmin);
        float px = fmaf(dx, t, ox);
        float py = fmaf(dy, t, oy);
        float pz = fmaf(dz, t, oz);

        bool mask = hit && (s < NSAMP) && (t < tmax)
                    && (fabsf(px) <= F_RADIUS)
                    && (fabsf(py) <= F_RADIUS)
                    && (fabsf(pz) <= F_RADIUS);

        // gfx1250: prefetch next block's corner records while this
        // block's gather + compositing is in flight (global_prefetch_b8)
        {
            float t2 = fmaf((float)(s + 32), F_STEP, tmin);
            if (hit && (s + 32 < NSAMP) && (t2 < tmax)) {
                float qx = fmaf(dx, t2, ox), qy = fmaf(dy, t2, oy), qz = fmaf(dz, t2, oz);
                float ffx, ffy, ffz; int jx, jy, jz;
                int pb = plx_coords(qx, qy, qz, ffx, ffy, ffz, jx, jy, jz);
                __builtin_prefetch(grid + pb, 0, 1);
                __builtin_prefetch(grid + pb + VZ_STRIDE, 0, 1);
                __builtin_prefetch(grid + pb + VY_STRIDE, 0, 1);
                __builtin_prefetch(grid + pb + VZ_STRIDE + VY_STRIDE, 0, 1);
            }
        }

        float alpha = 0.0f;
        float ch[DATA_DIM];
        if (mask) {
            float fx, fy, fz; int ix, iy, iz;
            int base = plx_coords(px, py, pz, fx, fy, fz, ix, iy, iz);

            float wx0 = 1.0f - fx, wx1 = fx;
            float wy0 = 1.0f - fy, wy1 = fy;
            float wz0 = 1.0f - fz, wz1 = fz;
            float w[8] = { wz0*wy0*wx0, wz0*wy0*wx1, wz0*wy1*wx0, wz0*wy1*wx1,
                           wz1*wy0*wx0, wz1*wy0*wx1, wz1*wy1*wx0, wz1*wy1*wx1 };
            const int off[8] = { 0,                         VX_STRIDE,
                                 VY_STRIDE,                 VY_STRIDE + VX_STRIDE,
                                 VZ_STRIDE,                 VZ_STRIDE + VX_STRIDE,
                                 VZ_STRIDE + VY_STRIDE,     VZ_STRIDE + VY_STRIDE + VX_STRIDE };

            float4 acc[7];
#pragma unroll
            for (int q = 0; q < 7; ++q) acc[q] = make_float4(0.f, 0.f, 0.f, 0.f);

#pragma unroll
            for (int j = 0; j < 8; ++j) {
                const float4* p = (const float4*)(grid + base + off[j]); // 16B aligned
                float wj = w[j];
#pragma unroll
                for (int q = 0; q < 7; ++q) {
                    float4 v = p[q];
                    acc[q].x = fmaf(wj, v.x, acc[q].x);
                    acc[q].y = fmaf(wj, v.y, acc[q].y);
                    acc[q].z = fmaf(wj, v.z, acc[q].z);
                    acc[q].w = fmaf(wj, v.w, acc[q].w);
                }
            }
#pragma unroll
            for (int q = 0; q < 7; ++q) {
                ch[4*q+0] = acc[q].x; ch[4*q+1] = acc[q].y;
                ch[4*q+2] = acc[q].z; ch[4*q+3] = acc[q].w;
            }
            float sg = fmaxf(ch[27], 0.0f);
            alpha = 1.0f - expf(-sg * (F_STEP * nd));
        }

        // ---- wave32 inclusive product scan of f = 1 - alpha + 1e-10 ----
        float f = 1.0f - alpha + 1e-10f;
        float scan = f;
#pragma unroll
        for (int o = 1; o < 32; o <<= 1) {
            float u = __shfl_up(scan, o, 32);
            if (lane >= o) scan *= u;
        }
        float Texcl = __shfl_up(scan, 1, 32);
        if (lane == 0) Texcl = 1.0f;
        float wgt = alpha * Tc * Texcl;          // abs_light for this sample
        Tc *= __shfl(scan, 31, 32);              // carry whole-block product

        if (mask) {
            float r0 = ch[0]*m0 + ch[1]*m1 + ch[2]*m2 + ch[3]*m3 + ch[4]*m4
                     + ch[5]*m5 + ch[6]*m6 + ch[7]*m7 + ch[8]*m8;
            float r1 = ch[ 9]*m0 + ch[10]*m1 + ch[11]*m2 + ch[12]*m3 + ch[13]*m4
                     + ch[14]*m5 + ch[15]*m6 + ch[16]*m7 + ch[17]*m8;
            float r2 = ch[18]*m0 + ch[19]*m1 + ch[20]*m2 + ch[21]*m3 + ch[22]*m4
                     + ch[23]*m5 + ch[24]*m6 + ch[25]*m7 + ch[26]*m8;
            aR = fmaf(wgt, plx_sigmoid(r0), aR);
            aG = fmaf(wgt, plx_sigmoid(r1), aG);
            aB = fmaf(wgt, plx_sigmoid(r2), aB);
        }
        aA += wgt;                               // wgt==0 when !mask
    }

    // ---- butterfly reduction across the wave ----
#pragma unroll
    for (int o = 16; o >= 1; o >>= 1) {
        aR += __shfl_xor(aR, o, 32);
        aG += __shfl_xor(aG, o, 32);
        aB += __shfl_xor(aB, o, 32);
        aA += __shfl_xor(aA, o, 32);
    }
    if (lane == 0) {
        float bg = 1.0f - aA;
        out[3*ray+0] = aR + bg;
        out[3*ray+1] = aG + bg;
        out[3*ray+2] = aB + bg;
    }
}

// ---------------------------------------------------------------
// Fallback: gather straight from the (C,Z,Y,X) layout if the
// workspace is too small for the transposed copy.
// ---------------------------------------------------------------
__global__ __launch_bounds__(128) void plx_render_direct_kernel(
    const float* __restrict__ ro, const float* __restrict__ rd,
    const float* __restrict__ data, float* __restrict__ out)
{
    const int lane = threadIdx.x & 31;
    const int ray  = blockIdx.x * 4 + (threadIdx.x >> 5);

    float ox = ro[3*ray+0], oy = ro[3*ray+1], oz = ro[3*ray+2];
    float dx = rd[3*ray+0], dy = rd[3*ray+1], dz = rd[3*ray+2];

    float ivx = 1.0f/dx, ivy = 1.0f/dy, ivz = 1.0f/dz;
    float t0x = (-F_RADIUS-ox)*ivx, t1x = (F_RADIUS-ox)*ivx;
    float t0y = (-F_RADIUS-oy)*ivy, t1y = (F_RADIUS-oy)*ivy;
    float t0z = (-F_RADIUS-oz)*ivz, t1z = (F_RADIUS-oz)*ivz;
    float tmin = fmaxf(fmaxf(fminf(t0x,t1x), fminf(t0y,t1y)), fminf(t0z,t1z));
    tmin = fmaxf(tmin, 0.0f);
    float tmax = fminf(fminf(fmaxf(t0x,t1x), fmaxf(t0y,t1y)), fmaxf(t0z,t1z));
    bool  hit  = (tmin < tmax);

    float nd  = sqrtf(dx*dx + dy*dy + dz*dz);
    float inn = 1.0f / (nd + 1e-9f);
    float nx = dx*inn, ny = dy*inn, nz = dz*inn;

    float m0 = 0.28209479177387814f;
    float m1 = -0.4886025119029199f * ny;
    float m2 =  0.4886025119029199f * nz;
    float m3 = -0.4886025119029199f * nx;
    float m4 =  1.0925484305920792f * nx * ny;
    float m5 = -1.0925484305920792f * ny * nz;
    float m6 =  0.31539156525252005f * (2.0f*nz*nz - nx*nx - ny*ny);
    float m7 = -1.0925484305920792f * nx * nz;
    float m8 =  0.5462742152960396f * (nx*nx - ny*ny);

    float Tc = 1.0f;
    float aR = 0.0f, aG = 0.0f, aB = 0.0f, aA = 0.0f;

    for (int blk = 0; blk < NBLK; ++blk) {
        int   s = blk * 32 + lane;
        float t = fmaf((float)s, F_STEP, tmin);
        float px = fmaf(dx, t, ox);
        float py = fmaf(dy, t, oy);
        float pz = fmaf(dz, t, oz);

        bool mask = hit && (s < NSAMP) && (t < tmax)
                    && (fabsf(px) <= F_RADIUS)
                    && (fabsf(py) <= F_RADIUS)
                    && (fabsf(pz) <= F_RADIUS);

        float alpha = 0.0f;
        float ch[DATA_DIM];
        if (mask) {
            float fx, fy, fz; int ix, iy, iz;
            const float S = 0.5f * 127.0f / F_RADIUS;
            float gx = fminf(fmaxf(fmaf(px, S, 63.5f), 0.0f), 127.0f);
            float gy = fminf(fmaxf(fmaf(py, S, 63.5f), 0.0f), 127.0f);
            float gz = fminf(fmaxf(fmaf(pz, S, 63.5f), 0.0f), 127.0f);
            ix = min((int)gx, 126); iy = min((int)gy, 126); iz = min((int)gz, 126);
            fx = gx - (float)ix; fy = gy - (float)iy; fz = gz - (float)iz;

            float wx0 = 1.0f - fx, wx1 = fx;
            float wy0 = 1.0f - fy, wy1 = fy;
            float wz0 = 1.0f - fz, wz1 = fz;
            float w[8] = { wz0*wy0*wx0, wz0*wy0*wx1, wz0*wy1*wx0, wz0*wy1*wx1,
                           wz1*wy0*wx0, wz1*wy0*wx1, wz1*wy1*wx0, wz1*wy1*wx1 };
            int i000 = (iz * RES + iy) * RES + ix;
            int idx[8] = { i000,               i000 + 1,
                           i000 + RES,         i000 + RES + 1,
                           i000 + RES*RES,     i000 + RES*RES + 1,
                           i000 + RES*RES+RES, i000 + RES*RES + RES + 1 };
#pragma unroll
            for (int c = 0; c < DATA_DIM; ++c) {
                const float* p = data + c * NV;
                float v = 0.0f;
#pragma unroll
                for (int j = 0; j < 8; ++j) v = fmaf(w[j], p[idx[j]], v);
                ch[c] = v;
            }
            float sg = fmaxf(ch[27], 0.0f);
            alpha = 1.0f - expf(-sg * (F_STEP * nd));
        }

        float f = 1.0f - alpha + 1e-10f;
        float scan = f;
#pragma unroll
        for (int o = 1; o < 32; o <<= 1) {
            float u = __shfl_up(scan, o, 32);
            if (lane >= o) scan *= u;
        }
        float Texcl = __shfl_up(scan, 1, 32);
        if (lane == 0) Texcl = 1.0f;
        float wgt = alpha * Tc * Texcl;
        Tc *= __shfl(scan, 31, 32);

        if (mask) {
            float r0 = ch[0]*m0 + ch[1]*m1 + ch[2]*m2 + ch[3]*m3 + ch[4]*m4
                     + ch[5]*m5 + ch[6]*m6 + ch[7]*m7 + ch[8]*m8;
            float r1 = ch[ 9]*m0 + ch[10]*m1 + ch[11]*m2 + ch[12]*m3 + ch[13]*m4
                     + ch[14]*m5 + ch[15]*m6 + ch[16]*m7 + ch[17]*m8;
            float r2 = ch[18]*m0 + ch[19]*m1 + ch[20]*m2 + ch[21]*m3 + ch[22]*m4
                     + ch[23]*m5 + ch[24]*m6 + ch[25]*m7 + ch[26]*m8;
            aR = fmaf(wgt, plx_sigmoid(r0), aR);
            aG = fmaf(wgt, plx_sigmoid(r1), aG);
            aB = fmaf(wgt, plx_sigmoid(r2), aB);
        }
        aA += wgt;
    }

#pragma unroll
    for (int o = 16; o >= 1; o >>= 1) {
        aR += __shfl_xor(aR, o, 32);
        aG += __shfl_xor(aG, o, 32);
        aB += __shfl_xor(aB, o, 32);
        aA += __shfl_xor(aA, o, 32);
    }
    if (lane == 0) {
        float bg = 1.0f - aA;
        out[3*ray+0] = aR + bg;
        out[3*ray+1] = aG + bg;
        out[3*ray+2] = aB + bg;
    }
}

extern "C" void kernel_launch(void* const* d_in, const int* in_sizes, int n_in,
                              void* d_out, int out_size, void* d_ws, size_t ws_size,
                              hipStream_t stream) {
    (void)in_sizes; (void)n_in; (void)out_size;
    const float* rays_o = (const float*)d_in[0];
    const float* rays_d = (const float*)d_in[1];
    const float* data   = (const float*)d_in[2];
    float* out = (float*)d_out;

    const size_t need = (size_t)NV * DATA_DIM * sizeof(float);   // ~235 MB
    if (ws_size >= need) {
        float* grid = (float*)d_ws;
        plx_transpose_tdm_kernel<<<NV / TP_VOX, 256, 0, stream>>>(data, grid);
        plx_render_tx_kernel<<<NRAYS / 4, 128, 0, stream>>>(rays_o, rays_d, grid, out);
    } else {
        plx_render_direct_kernel<<<NRAYS / 4, 128, 0, stream>>>(rays_o, rays_d, data, out);
    }
}